// EulerIntegration_7060926235109
// MI455X (gfx1250) — compile-verified
//
#include <hip/hip_runtime.h>
#include <hip/hip_bf16.h>

// Problem constants (fixed by the reference).
static constexpr int EI_W  = 1920;
static constexpr int EI_H  = 1080;
static constexpr int EI_HW = EI_W * EI_H;   // 2,073,600 ; divisible by 256 (8100 blocks)

// gfx1250 async global->LDS path, guarded so a missing builtin can't break the build.
#if defined(__AMDGCN__) && __has_builtin(__builtin_amdgcn_global_load_async_to_lds_b32) && \
    __has_builtin(__builtin_amdgcn_s_wait_asynccnt)
#define EI_ASYNC 1
#else
#define EI_ASYNC 0
#endif

#if EI_ASYNC
typedef __attribute__((address_space(1))) int ei_gint;   // global int
typedef __attribute__((address_space(3))) int ei_lint;   // LDS int
#endif

// ---------------------------------------------------------------------------
// Pass 1 (optional): interleave planar motion [B,2,H,W] -> packed [B,H*W] float2
// so the integration loop's data-dependent gather is a single global_load_b64.
// Uses the CDNA5 async global->LDS load (ASYNCcnt) when available.
// ---------------------------------------------------------------------------
__global__ __launch_bounds__(256) void ei_interleave_kernel(const float* __restrict__ motion,
                                                            float2* __restrict__ mpk) {
  const int b = blockIdx.y;
  const int i = blockIdx.x * 256 + (int)threadIdx.x;            // exact tiling, no bounds check
  const float* __restrict__ p0 = motion + (size_t)b * (2 * (size_t)EI_HW) + i;  // dx plane
  const float* __restrict__ p1 = p0 + EI_HW;                                     // dy plane
  float vx, vy;
#if EI_ASYNC
  __shared__ float s0[256];
  __shared__ float s1[256];
  // global -> LDS async copies, tracked by ASYNCcnt (gfx1250 path).
  __builtin_amdgcn_global_load_async_to_lds_b32(
      (ei_gint*)p0, (ei_lint*)&s0[threadIdx.x], 0, 0);
  __builtin_amdgcn_global_load_async_to_lds_b32(
      (ei_gint*)p1, (ei_lint*)&s1[threadIdx.x], 0, 0);
  __builtin_amdgcn_s_wait_asynccnt(0);
  // Each lane reads back only its own slot -> no workgroup barrier needed.
  vx = s0[threadIdx.x];
  vy = s1[threadIdx.x];
#else
  // Streaming, read-once: non-temporal so we don't pollute L2 ahead of the hot pass.
  vx = __builtin_nontemporal_load(p0);
  vy = __builtin_nontemporal_load(p1);
#endif
  mpk[(size_t)b * EI_HW + i] = make_float2(vx, vy);
}

// ---------------------------------------------------------------------------
// Pass 2: Euler integration. One thread per pixel per batch.
// Key transforms vs reference (bit-exact):
//  * gather fusion: step t's "second" gather == step t+1's "first" gather
//    -> exactly one gather per active step.
//  * invalidation early-exit: once OOB, every remaining step adds
//    motion[:,y,x]; replay as repeated float adds (matches scan rounding).
// ---------------------------------------------------------------------------
template <bool PACKED>
__global__ __launch_bounds__(256) void ei_integrate_kernel(const float*  __restrict__ motion,
                                                           const float2* __restrict__ mpk,
                                                           const int*    __restrict__ dest,
                                                           float*        __restrict__ out) {
  const int b = blockIdx.y;
  const int p = blockIdx.x * 256 + (int)threadIdx.x;
  const int x = p % EI_W;
  const int y = p / EI_W;
  const float fx = (float)x;
  const float fy = (float)y;
  const int nsteps = dest[b];                       // uniform per block -> scalar load

  const float2* __restrict__ mp = mpk ? (mpk + (size_t)b * EI_HW) : nullptr;
  const float*  __restrict__ m0 = motion + (size_t)b * (2 * (size_t)EI_HW);
  const float*  __restrict__ m1 = m0 + EI_HW;

  // Motion at the identity coordinate (== first gather, since round(coord)==coord).
  float idx_mx, idx_my;
  if constexpr (PACKED) {
    const float2 g = mp[p];
    idx_mx = g.x; idx_my = g.y;
  } else {
    idx_mx = m0[p]; idx_my = m1[p];
  }

  float dcx = fx, dcy = fy;       // clamped/reset coordinate track
  float ux  = fx, uy  = fy;       // unclamped accumulator track
  float gx = idx_mx, gy = idx_my; // motion at round(dc) for the *current* step

  int t = 1;
  for (; t <= nsteps; ++t) {
    const float cx = dcx + gx;
    const float cy = dcy + gy;
    if (cx > (float)(EI_W - 1) || cx < 0.0f || cy > (float)(EI_H - 1) || cy < 0.0f)
      break;                                        // pixel invalidates at step t
    dcx = cx; dcy = cy;
    const int ix = (int)__builtin_rintf(dcx);       // v_rndne_f32: round-half-to-even == jnp.round
    const int iy = (int)__builtin_rintf(dcy);
    const int q  = iy * EI_W + ix;                  // in-bounds by construction
    if constexpr (PACKED) {
      const float2 g = mp[q];                       // one global_load_b64 gather per step
      gx = g.x; gy = g.y;
    } else {
      gx = m0[q]; gy = m1[q];                       // planar fallback: two b32 gathers
    }
    ux += gx; uy += gy;                             // second-gather accumulation (fused)
  }
  // Invalidated tail: steps t..nsteps each add motion at identity. Repeated adds
  // (not multiply) to reproduce the reference scan's rounding exactly.
  for (; t <= nsteps; ++t) { ux += idx_mx; uy += idx_my; }

  // displacement = dcu - coord ; streaming output -> non-temporal stores so the
  // 66 MB result doesn't evict the L2-resident motion field.
  float* __restrict__ o0 = out + (size_t)b * (2 * (size_t)EI_HW);
  __builtin_nontemporal_store(ux - fx, o0 + p);
  __builtin_nontemporal_store(uy - fy, o0 + EI_HW + p);
}

// ---------------------------------------------------------------------------
extern "C" void kernel_launch(void* const* d_in, const int* in_sizes, int n_in,
                              void* d_out, int out_size, void* d_ws, size_t ws_size,
                              hipStream_t stream) {
  const float* motion = (const float*)d_in[0];        // [B,2,H,W] fp32
  const int*   dest   = (const int*)d_in[1];          // [B] int32
  float*       out    = (float*)d_out;                // [B,2,H,W] fp32
  const int B = (n_in > 1) ? in_sizes[1] : 4;

  dim3 grid(EI_HW / 256, B);
  const size_t need = (size_t)B * (size_t)EI_HW * sizeof(float2);

  if (d_ws != nullptr && ws_size >= need) {
    // Pass 1: build packed float2 motion in workspace; Pass 2: integrate with b64 gathers.
    ei_interleave_kernel<<<grid, 256, 0, stream>>>(motion, (float2*)d_ws);
    ei_integrate_kernel<true><<<grid, 256, 0, stream>>>(motion, (const float2*)d_ws, dest, out);
  } else {
    // Workspace too small: planar-gather fallback (still one fused gather pair per step).
    ei_integrate_kernel<false><<<grid, 256, 0, stream>>>(motion, nullptr, dest, out);
  }
}